// EditorUnembedCrossAttention_53695681135065
// MI455X (gfx1250) — compile-verified
//
#include <hip/hip_runtime.h>

typedef __attribute__((ext_vector_type(2))) float v2f;
typedef __attribute__((ext_vector_type(4))) float v4f;
typedef __attribute__((ext_vector_type(8))) float v8f;

#define DD   1024      // EMBED_DIM
#define BB   8         // batch
#define QQ   4096      // query length
#define SS   128       // hidden_states seq len
// MULT = 2, HEAD_DIM = 64, HPM = 16 (softmax over singleton axis => weights == 1,
// so q/scores are dead code; output = broadcast of the v->proj row per batch).

// ---------------------------------------------------------------------------
// Stage A: v[m][b][n] = sum_k h_last[b][k] * c_attn_w[m][k][D + n] + c_attn_b[m][D+n]
// One wave per 16x16 output tile (M padded 8->16, N tile of 16), K-loop with
// V_WMMA_F32_16X16X4_F32. 2 m-slices * 64 N-tiles = 128 waves total.
// Padded rows are produced branchlessly: load a valid aliased row, multiply by
// a 0/1 lane mask (finite * 0 == 0), keeping EXEC untouched in the hot loop.
// ---------------------------------------------------------------------------
__global__ void __launch_bounds__(256)
stageA_kv_v(const float* __restrict__ hs,     // (8,128,1024)
            const float* __restrict__ caw,    // (2,1024,2048)
            const float* __restrict__ cab,    // (2,2048)
            float* __restrict__ v_ws)         // (2,8,1024)
{
    const int wave = (blockIdx.x * blockDim.x + threadIdx.x) >> 5;   // 0..127
    const int lane = threadIdx.x & 31;
    const int m    = wave >> 6;                 // 0..1
    const int n0   = (wave & 63) << 4;          // column tile base
    const int half = lane >> 4;                 // 0 or 1
    const int r    = lane & 15;                 // row / column within tile

    // A row pointer: h_last[b][:] = hs[b, S-1, :]; rows 8..15 alias b = r-8,
    // zeroed via mask so no divergent branch in the loop.
    const float mask = (r < 8) ? 1.0f : 0.0f;
    const float* hrow = hs + (size_t)(r & 7) * (SS * DD) + (size_t)(SS - 1) * DD;
    // B column pointer: c_attn_w[m][:, D + n0 + r]
    const float* bcol = caw + (size_t)m * (DD * 2 * DD) + DD + n0 + r;

    v8f acc = {0.f, 0.f, 0.f, 0.f, 0.f, 0.f, 0.f, 0.f};
    #pragma unroll 4
    for (int k0 = 0; k0 < DD; k0 += 4) {
        const int ka = k0 + 2 * half;
        const v2f araw = *(const v2f*)(hrow + ka);   // unconditional b64 load
        v2f a;
        a.x = araw.x * mask;
        a.y = araw.y * mask;
        v2f bf;
        bf.x = bcol[(size_t)ka * (2 * DD)];
        bf.y = bcol[(size_t)(ka + 1) * (2 * DD)];
        acc = __builtin_amdgcn_wmma_f32_16x16x4_f32(
                  false, a, false, bf, (short)0, acc, false, false);
    }

    if (half == 0) {            // lanes 0..15, VGPR j == row b=j (j<8 real)
        const float bias = cab[(size_t)m * (2 * DD) + DD + n0 + r];
        #pragma unroll
        for (int j = 0; j < 8; ++j) {
            v_ws[(size_t)m * (BB * DD) + (size_t)j * DD + n0 + r] = acc[j] + bias;
        }
    }
}

// ---------------------------------------------------------------------------
// Stage B: y[m][b][n] = sum_k v[m][b][k] * c_proj_w[m][k][n] + c_proj_b[m][n]
// Identical tiling to stage A.
// ---------------------------------------------------------------------------
__global__ void __launch_bounds__(256)
stageB_proj(const float* __restrict__ v_ws,   // (2,8,1024)
            const float* __restrict__ cpw,    // (2,1024,1024)
            const float* __restrict__ cpb,    // (2,1024)
            float* __restrict__ y_ws)         // (2,8,1024)
{
    const int wave = (blockIdx.x * blockDim.x + threadIdx.x) >> 5;   // 0..127
    const int lane = threadIdx.x & 31;
    const int m    = wave >> 6;
    const int n0   = (wave & 63) << 4;
    const int half = lane >> 4;
    const int r    = lane & 15;

    const float mask = (r < 8) ? 1.0f : 0.0f;
    const float* arow = v_ws + (size_t)m * (BB * DD) + (size_t)(r & 7) * DD;
    const float* bcol = cpw + (size_t)m * (DD * DD) + n0 + r;

    v8f acc = {0.f, 0.f, 0.f, 0.f, 0.f, 0.f, 0.f, 0.f};
    #pragma unroll 4
    for (int k0 = 0; k0 < DD; k0 += 4) {
        const int ka = k0 + 2 * half;
        const v2f araw = *(const v2f*)(arow + ka);
        v2f a;
        a.x = araw.x * mask;
        a.y = araw.y * mask;
        v2f bf;
        bf.x = bcol[(size_t)ka * DD];
        bf.y = bcol[(size_t)(ka + 1) * DD];
        acc = __builtin_amdgcn_wmma_f32_16x16x4_f32(
                  false, a, false, bf, (short)0, acc, false, false);
    }

    if (half == 0) {
        const float bias = cpb[(size_t)m * DD + n0 + r];
        #pragma unroll
        for (int j = 0; j < 8; ++j) {
            y_ws[(size_t)m * (BB * DD) + (size_t)j * DD + n0 + r] = acc[j] + bias;
        }
    }
}

// ---------------------------------------------------------------------------
// Broadcast: out[b][q][:] = y[0][b][:] + (b==0 ? y[1][0][:] : 0), for all q.
// 134 MB of write-once output -> non-temporal b128 stores. One block per
// (b, 16-row q chunk): each thread keeps one float4 of the row in registers
// and streams it to 16 q rows. This is the HBM-bound part (~5.8 us floor
// at 23.3 TB/s).
// ---------------------------------------------------------------------------
__global__ void __launch_bounds__(256)
bcast_out(const float* __restrict__ y_ws, float* __restrict__ out)
{
    const int b     = blockIdx.x >> 8;            // 0..7
    const int qBase = (blockIdx.x & 255) << 4;    // 0..4080 step 16
    const int t     = threadIdx.x;                // float4 index 0..255

    const v4f* y0 = (const v4f*)(y_ws + (size_t)b * DD);
    v4f val = y0[t];
    if (b == 0) {
        const v4f* y1 = (const v4f*)(y_ws + (size_t)BB * DD);   // m=1, b=0 row
        val += y1[t];
    }

    v4f* outp = (v4f*)(out + ((size_t)b * QQ + qBase) * DD);
    #pragma unroll
    for (int q = 0; q < 16; ++q) {
        __builtin_nontemporal_store(val, outp + (size_t)q * (DD / 4) + t);
    }
}

// ---------------------------------------------------------------------------
extern "C" void kernel_launch(void* const* d_in, const int* in_sizes, int n_in,
                              void* d_out, int out_size, void* d_ws, size_t ws_size,
                              hipStream_t stream) {
    // setup_inputs order:
    // 0: encoder_hidden_states (dead: softmax over singleton axis == 1)
    // 1: hidden_states  2: q_w (dead)  3: q_b (dead)
    // 4: c_attn_w  5: c_attn_b  6: c_proj_w  7: c_proj_b
    const float* hs  = (const float*)d_in[1];
    const float* caw = (const float*)d_in[4];
    const float* cab = (const float*)d_in[5];
    const float* cpw = (const float*)d_in[6];
    const float* cpb = (const float*)d_in[7];
    float* out = (float*)d_out;

    float* v_ws = (float*)d_ws;                 // 2*8*1024 floats
    float* y_ws = v_ws + 2 * BB * DD;           // 2*8*1024 floats

    // 128 waves each => 16 blocks x 8 waves (wave32)
    stageA_kv_v<<<16, 256, 0, stream>>>(hs, caw, cab, v_ws);
    stageB_proj<<<16, 256, 0, stream>>>(v_ws, cpw, cpb, y_ws);
    // 8 b-slices * 256 q-chunks = 2048 blocks
    bcast_out<<<2048, 256, 0, stream>>>(y_ws, out);
}